// FederatedCausalGNN_59768764891878
// MI455X (gfx1250) — compile-verified
//
#include <hip/hip_runtime.h>
#include <math.h>

// ---------------------------------------------------------------------------
// FederatedCausalGNN forward for MI455X (gfx1250, wave32, WMMA)
//   3x GCNConv (D^-1/2 (A+I) D^-1/2 (hW) + b) + two MLP heads w/ sigmoid
// Memory-bound on the edge scatter (~5 GB -> ~0.25 ms @ 23.3 TB/s);
// GEMMs done with V_WMMA_F32_16X16X4_F32 (f32-exact), A tiles staged in LDS.
// ---------------------------------------------------------------------------

typedef __attribute__((ext_vector_type(2))) float v2f;
typedef __attribute__((ext_vector_type(8))) float v8f;

#define NN 100000
#define EE 1600000
#define FF 256
#define HH 128

static_assert(NN % 16 == 0, "row tiles must divide N");

// ------------------------------- utilities --------------------------------

__global__ void fill_f32(float* __restrict__ p, float v, size_t n) {
  size_t i = (size_t)blockIdx.x * blockDim.x + threadIdx.x;
  if (i < n) p[i] = v;
}

__global__ void deg_accum(const int* __restrict__ dst, float* __restrict__ deg) {
  int e = blockIdx.x * blockDim.x + threadIdx.x;
  if (e < EE) atomicAdd(&deg[dst[e]], 1.0f);
}

__global__ void dinv_finalize(float* __restrict__ deg_dinv) {
  int i = blockIdx.x * blockDim.x + threadIdx.x;
  if (i < NN) deg_dinv[i] = rsqrtf(deg_dinv[i] + 1.0f);
}

// ------------------------------ WMMA GEMM ---------------------------------
// out[M x NCOLS] = A[M x K] @ W[K x NCOLS]   (optionally + bias, ReLU)
// One block per 16-row tile; wave w owns output cols [16w, 16w+16).
// A tile staged in LDS with padded stride (K+4 floats) -> conflict-free
// ds_load_b64 fragment reads (banks = (4*idx + 2*half) % 64, all distinct).

template <int K, int NCOLS, bool BIAS_RELU>
__global__ __launch_bounds__((NCOLS / 16) * 32)
void gemm_wmma_f32(const float* __restrict__ A, const float* __restrict__ W,
                   const float* __restrict__ bias, float* __restrict__ out) {
  constexpr int NTHREADS = (NCOLS / 16) * 32;
  constexpr int LSTR = K + 4;  // padded LDS row stride (floats)
  __shared__ __align__(16) float As[16 * LSTR];

  const int tid = threadIdx.x;
  const int m0 = blockIdx.x * 16;

  // Cooperative, coalesced b128 stage of the 16 x K A tile.
  for (int t = tid; t < 16 * (K / 4); t += NTHREADS) {
    const int r = t / (K / 4);
    const int c = (t % (K / 4)) * 4;
    const float4 v = *(const float4*)(A + (size_t)(m0 + r) * K + c);
    *(float4*)(&As[r * LSTR + c]) = v;
  }
  __syncthreads();

  const int lane = tid & 31;
  const int wave = tid >> 5;
  const int idx  = lane & 15;   // row (A) / col (B,C) within tile
  const int half = lane >> 4;   // K-pair selector per ISA 16x4 layout
  const int n0   = wave * 16;

  const float* arow = &As[idx * LSTR];
  v8f acc = {};

#pragma unroll 8
  for (int k = 0; k < K; k += 4) {
    v2f a, b;
    // A 16x4 f32 layout: lanes 0-15 -> K={k,k+1}, lanes 16-31 -> K={k+2,k+3}
    a.x = arow[k + 2 * half];
    a.y = arow[k + 2 * half + 1];
    // B 4x16 f32 layout: VGPR0 -> K = k+2*half, VGPR1 -> K = k+2*half+1
    b.x = W[(size_t)(k + 2 * half) * NCOLS + n0 + idx];
    b.y = W[(size_t)(k + 2 * half + 1) * NCOLS + n0 + idx];
    acc = __builtin_amdgcn_wmma_f32_16x16x4_f32(
        /*neg_a=*/false, a, /*neg_b=*/false, b,
        /*c_mod=*/(short)0, acc, /*reuse_a=*/false, /*reuse_b=*/false);
  }

  // C/D 16x16 f32 layout: VGPR r -> M = r + 8*half, N = idx
  const int col = n0 + idx;
  float bval = 0.0f;
  if (BIAS_RELU) bval = bias[col];
#pragma unroll
  for (int r = 0; r < 8; ++r) {
    const int row = m0 + r + 8 * half;
    float v = acc[r];
    if (BIAS_RELU) v = fmaxf(v + bval, 0.0f);
    out[(size_t)row * NCOLS + col] = v;
  }
}

// --------------------------- edge aggregation -----------------------------
// One wave per edge; lane l moves features [4l, 4l+4) (coalesced 512B row
// read) scaled by dinv[src]*dinv[dst], atomically accumulated into agg[dst].

__global__ void edge_aggregate(const int* __restrict__ src,
                               const int* __restrict__ dst,
                               const float* __restrict__ dinv,
                               const float* __restrict__ hw,
                               float* __restrict__ agg) {
  const int g = blockIdx.x * blockDim.x + threadIdx.x;
  const int e = g >> 5;
  const int lane = g & 31;
  if (e >= EE) return;
  const int s = src[e];
  const int d = dst[e];
  const float coef = dinv[s] * dinv[d];
  const float4 v = *(const float4*)(hw + (size_t)s * HH + lane * 4);
  float* ag = agg + (size_t)d * HH + lane * 4;
  atomicAdd(ag + 0, v.x * coef);
  atomicAdd(ag + 1, v.y * coef);
  atomicAdd(ag + 2, v.z * coef);
  atomicAdd(ag + 3, v.w * coef);
}

// ---------------------- self-loop + bias (+ ReLU) -------------------------

template <bool RELU>
__global__ void selfloop_bias_act(const float* __restrict__ agg,
                                  const float* __restrict__ hw,
                                  const float* __restrict__ dinv,
                                  const float* __restrict__ bias,
                                  float* __restrict__ out) {
  const size_t i = (size_t)blockIdx.x * blockDim.x + threadIdx.x;
  if (i >= (size_t)NN * HH) return;
  const int row = (int)(i >> 7);  // / 128
  const int col = (int)(i & 127);
  const float di = dinv[row];
  float v = agg[i] + hw[i] * di * di + bias[col];
  if (RELU) v = fmaxf(v, 0.0f);
  out[i] = v;
}

// ------------------------- head output (64 -> 1) --------------------------
// One wave per node: lane l holds hidden [2l, 2l+2), wave-reduce dot, sigmoid.

__global__ void head_out(const float* __restrict__ hid,
                         const float* __restrict__ w2,
                         const float* __restrict__ b2,
                         float* __restrict__ out) {
  const int row = blockIdx.x * (blockDim.x >> 5) + (threadIdx.x >> 5);
  const int lane = threadIdx.x & 31;
  if (row >= NN) return;
  const float2 h = *(const float2*)(hid + (size_t)row * 64 + lane * 2);
  const float2 w = *(const float2*)(w2 + lane * 2);
  float p = h.x * w.x + h.y * w.y;
#pragma unroll
  for (int off = 16; off > 0; off >>= 1) p += __shfl_down(p, off);
  if (lane == 0) out[row] = 1.0f / (1.0f + expf(-(p + b2[0])));
}

// ------------------------------- launcher ---------------------------------

extern "C" void kernel_launch(void* const* d_in, const int* in_sizes, int n_in,
                              void* d_out, int out_size, void* d_ws, size_t ws_size,
                              hipStream_t stream) {
  (void)in_sizes; (void)n_in; (void)out_size; (void)ws_size;

  const float* x   = (const float*)d_in[0];
  const int*   ei  = (const int*)d_in[1];
  const float* W1  = (const float*)d_in[2];  const float* b1  = (const float*)d_in[3];
  const float* W2  = (const float*)d_in[4];  const float* b2  = (const float*)d_in[5];
  const float* W3  = (const float*)d_in[6];  const float* b3  = (const float*)d_in[7];
  const float* Wc1 = (const float*)d_in[8];  const float* bc1 = (const float*)d_in[9];
  const float* Wc2 = (const float*)d_in[10]; const float* bc2 = (const float*)d_in[11];
  const float* Wr1 = (const float*)d_in[12]; const float* br1 = (const float*)d_in[13];
  const float* Wr2 = (const float*)d_in[14]; const float* br2 = (const float*)d_in[15];

  const int* src = ei;        // edge_index[0]
  const int* dst = ei + EE;   // edge_index[1]

  // Workspace partition (floats): dinv | hw | agg | h
  float* ws   = (float*)d_ws;
  float* dinv = ws;                              // N      (deg, then rsqrt in place)
  float* hw   = ws + 102400;                     // N*H
  float* agg  = hw + (size_t)NN * HH;            // N*H
  float* hbuf = agg + (size_t)NN * HH;           // N*H (heads reuse as N*64)

  float* emb    = (float*)d_out;                 // [N,128]
  float* causal = emb + (size_t)NN * HH;         // [N]
  float* risk   = causal + NN;                   // [N]

  const size_t NH = (size_t)NN * HH;
  const int MT = NN / 16;                        // 6250 row tiles
  const int GN  = (NN + 255) / 256;
  const int GNH = (int)((NH + 255) / 256);
  const int GE  = (EE + 255) / 256;
  const int GEW = (EE * 32) / 256;               // one wave per edge
  const int GHD = NN / 8;                        // 8 rows per 256-thread block

  // ---- degree / dinv ----
  fill_f32<<<GN, 256, 0, stream>>>(dinv, 0.0f, (size_t)NN);
  deg_accum<<<GE, 256, 0, stream>>>(dst, dinv);
  dinv_finalize<<<GN, 256, 0, stream>>>(dinv);

  // ---- layer 1: F=256 -> H=128, ReLU ----
  gemm_wmma_f32<FF, HH, false><<<MT, 256, 0, stream>>>(x, W1, nullptr, hw);
  fill_f32<<<GNH, 256, 0, stream>>>(agg, 0.0f, NH);
  edge_aggregate<<<GEW, 256, 0, stream>>>(src, dst, dinv, hw, agg);
  selfloop_bias_act<true><<<GNH, 256, 0, stream>>>(agg, hw, dinv, b1, hbuf);

  // ---- layer 2: 128 -> 128, ReLU ----
  gemm_wmma_f32<HH, HH, false><<<MT, 256, 0, stream>>>(hbuf, W2, nullptr, hw);
  fill_f32<<<GNH, 256, 0, stream>>>(agg, 0.0f, NH);
  edge_aggregate<<<GEW, 256, 0, stream>>>(src, dst, dinv, hw, agg);
  selfloop_bias_act<true><<<GNH, 256, 0, stream>>>(agg, hw, dinv, b2, hbuf);

  // ---- layer 3: 128 -> 128, no activation -> emb (d_out) ----
  gemm_wmma_f32<HH, HH, false><<<MT, 256, 0, stream>>>(hbuf, W3, nullptr, hw);
  fill_f32<<<GNH, 256, 0, stream>>>(agg, 0.0f, NH);
  edge_aggregate<<<GEW, 256, 0, stream>>>(src, dst, dinv, hw, agg);
  selfloop_bias_act<false><<<GNH, 256, 0, stream>>>(agg, hw, dinv, b3, emb);

  // ---- causal head: relu(emb@Wc1+bc1) @ Wc2 + bc2 -> sigmoid ----
  gemm_wmma_f32<HH, 64, true><<<MT, 128, 0, stream>>>(emb, Wc1, bc1, hbuf);
  head_out<<<GHD, 256, 0, stream>>>(hbuf, Wc2, bc2, causal);

  // ---- risk head ----
  gemm_wmma_f32<HH, 64, true><<<MT, 128, 0, stream>>>(emb, Wr1, br1, hbuf);
  head_out<<<GHD, 256, 0, stream>>>(hbuf, Wr2, br2, risk);
}